// DGCNNEncoderGn_39075612459158
// MI455X (gfx1250) — compile-verified
//
#include <hip/hip_runtime.h>

// MI455X / gfx1250 implementation.
// Edge conv restructured:  h = (nbr-c)@wA^T + c@wB^T = Q[idx] + P
//   with Q = X^T wA^T,  P = X^T (wB-wA)^T;  max over k commutes with the
//   monotone affine+LeakyReLU, so each EdgeConv = Gram(WMMA f32) -> top-40
//   (LDS argmax) -> P/Q GEMM (WMMA f32) -> gather-max + activation.
// Final layer: fused WMMA GEMM (8-way A reuse) + BN + ReLU + max-over-N via
// uint atomicMax (valid: ReLU output >= 0).
// All GEMM kernels are templated on compile-time C so the K-dim guards fold
// away (branch-free WMMA inner loops); only C=3 keeps lane-selects.
// Workspace: ~51 MiB (xx 64KB, dist 16MiB per-batch, idx 2.5MiB, P/Q 16MiB each).

#define NPTS   2048
#define BATCH  8
#define KNN    40
#define CH_TOT 448
#define BN_INVF 0.9999950000374997f
#define SLOPEF  0.2f

typedef float v2f __attribute__((ext_vector_type(2)));
typedef float v8f __attribute__((ext_vector_type(8)));

#define V8F_ZERO {0.f,0.f,0.f,0.f,0.f,0.f,0.f,0.f}

// D = A(16x4) * B(4x16) + C, fp32, wave32.
// A: lanes 0-15 M=lane {K0,K1}, lanes 16-31 M=lane-16 {K2,K3}
// B: lanes 0-15 N=lane {K0,K1}, lanes 16-31 N=lane-16 {K2,K3}
// C/D: VGPR r, lanes 0-15 -> M=r, lanes 16-31 -> M=r+8, N=lane%16
__device__ __forceinline__ v8f wmma_f32_16x16x4(v2f a, v2f b, v8f c) {
#if __has_builtin(__builtin_amdgcn_wmma_f32_16x16x4_f32)
  return __builtin_amdgcn_wmma_f32_16x16x4_f32(
      /*neg_a=*/false, a, /*neg_b=*/false, b,
      /*c_mod=*/(short)0, c, /*reuse_a=*/false, /*reuse_b=*/false);
#else
  const int lane = threadIdx.x & 31;
  const int n = lane & 15;
  #pragma unroll
  for (int r = 0; r < 8; ++r) {
    const int M = r + ((lane >> 4) << 3);
    float s = c[r];
    #pragma unroll
    for (int k = 0; k < 4; ++k) {
      float av = __shfl((k & 1) ? a.y : a.x, M + ((k >> 1) << 4), 32);
      float bv = __shfl((k & 1) ? b.y : b.x, n + ((k >> 1) << 4), 32);
      s = fmaf(av, bv, s);
    }
    c[r] = s;
  }
  return c;
#endif
}

// X^T fragment loader: rows base..base+15, K-slice c0..c0+3 (C compile-time,
// so the guards fold away whenever C % 4 == 0 and c0+4 <= C).
template <int C>
__device__ __forceinline__ v2f load_xt_frag(const float* __restrict__ xb,
                                            int base, int m, int half, int c0) {
  const int k0 = c0 + 2 * half;
  v2f a;
  a.x = (k0     < C) ? xb[(size_t)k0       * NPTS + base + m] : 0.f;
  a.y = (k0 + 1 < C) ? xb[(size_t)(k0 + 1) * NPTS + base + m] : 0.f;
  return a;
}

// ---------------------------------------------------------------- squared norms
__global__ void xx_kernel(const float* __restrict__ x, long long bstride, int C,
                          float* __restrict__ xx) {
  int t = blockIdx.x * blockDim.x + threadIdx.x;      // over B*N
  int b = t / NPTS, n = t - b * NPTS;
  const float* xb = x + (size_t)b * (size_t)bstride;
  float s = 0.f;
  for (int c = 0; c < C; ++c) {
    float v = xb[(size_t)c * NPTS + n];
    s = fmaf(v, v, s);
  }
  xx[t] = s;
}

// -------------------------------------------------- dist tile (one batch) via WMMA
template <int C>
__global__ void gram_kernel(const float* __restrict__ xb, const float* __restrict__ xxb,
                            float* __restrict__ dist) {
  const int lane = threadIdx.x;
  const int i = blockIdx.x * 16;   // row block
  const int j = blockIdx.y * 16;   // col block
  const int m = lane & 15;
  const int half = (lane >> 4) & 1;
  v8f acc = V8F_ZERO;
  #pragma unroll
  for (int c0 = 0; c0 < C; c0 += 4) {
    v2f a  = load_xt_frag<C>(xb, i, m, half, c0);
    v2f bb = load_xt_frag<C>(xb, j, m, half, c0);
    acc = wmma_f32_16x16x4(a, bb, acc);
  }
  const int rbase = half * 8;
  const float xxc = xxb[j + m];
  #pragma unroll
  for (int r = 0; r < 8; ++r) {
    const int row = rbase + r;
    float d = 2.f * acc[r] - xxb[i + row] - xxc;
    dist[(size_t)(i + row) * NPTS + (j + m)] = d;
  }
}

// ------------------------------------------- top-40 per row: one wave32 per row
__global__ void topk_kernel(const float* __restrict__ dist, int* __restrict__ idxout) {
  __shared__ float row[NPTS];
  const int n = blockIdx.x;
  const int lane = threadIdx.x;
  for (int t = lane; t < NPTS; t += 32) row[t] = dist[(size_t)n * NPTS + t];
  __syncthreads();
  for (int kk = 0; kk < KNN; ++kk) {
    float bv = -3.4e38f;
    int bi = 0;
    for (int t = lane; t < NPTS; t += 32) {
      float v = row[t];
      if (v > bv) { bv = v; bi = t; }
    }
    #pragma unroll
    for (int off = 16; off > 0; off >>= 1) {
      float ov = __shfl_xor(bv, off, 32);
      int   oi = __shfl_xor(bi, off, 32);
      if (ov > bv || (ov == bv && oi < bi)) { bv = ov; bi = oi; }
    }
    if (lane == 0) {
      idxout[(size_t)n * KNN + kk] = bi;
      row[bi] = -3.4e38f;   // mask winner
    }
    __syncthreads();
  }
}

// ---------------- P = X^T (wB - wA)^T, Q = X^T wA^T  (all batches) via WMMA
template <int C, int O>
__global__ void pq_kernel(const float* __restrict__ x, long long bstride,
                          const float* __restrict__ w,
                          float* __restrict__ P, float* __restrict__ Q) {
  const int lane = threadIdx.x;
  const int b  = blockIdx.z;
  const int n0 = blockIdx.x * 16;
  const int o0 = blockIdx.y * 16;
  const float* xb = x + (size_t)b * (size_t)bstride;
  const int m = lane & 15;
  const int half = (lane >> 4) & 1;
  v8f accQ = V8F_ZERO;
  v8f accP = V8F_ZERO;
  #pragma unroll
  for (int c0 = 0; c0 < C; c0 += 4) {
    const int k0 = c0 + 2 * half;
    v2f a = load_xt_frag<C>(xb, n0, m, half, c0);
    const float wa0 = (k0     < C) ? w[(size_t)(o0 + m) * (2 * C) + k0    ] : 0.f;
    const float wa1 = (k0 + 1 < C) ? w[(size_t)(o0 + m) * (2 * C) + k0 + 1] : 0.f;
    const float wb0 = (k0     < C) ? w[(size_t)(o0 + m) * (2 * C) + C + k0    ] : 0.f;
    const float wb1 = (k0 + 1 < C) ? w[(size_t)(o0 + m) * (2 * C) + C + k0 + 1] : 0.f;
    v2f ba, bp;
    ba.x = wa0;        ba.y = wa1;
    bp.x = wb0 - wa0;  bp.y = wb1 - wa1;
    accQ = wmma_f32_16x16x4(a, ba, accQ);
    accP = wmma_f32_16x16x4(a, bp, accP);
  }
  const int rbase = half * 8;
  #pragma unroll
  for (int r = 0; r < 8; ++r) {
    const size_t off = ((size_t)b * NPTS + (n0 + rbase + r)) * O + o0 + m;
    Q[off] = accQ[r];
    P[off] = accP[r];
  }
}

// -------------- gather-max over neighbors + BN + LeakyReLU, write x_features
template <int O>
__global__ void edge_max_kernel(const float* __restrict__ P, const float* __restrict__ Q,
                                const int* __restrict__ idx,
                                const float* __restrict__ gamma, const float* __restrict__ beta,
                                float* __restrict__ xf, int chan_off) {
  __shared__ int sidx[KNN];
  const int n = blockIdx.x;
  const int b = blockIdx.y;
  for (int k = threadIdx.x; k < KNN; k += blockDim.x)
    sidx[k] = idx[((size_t)b * NPTS + n) * KNN + k];
  __syncthreads();
  const int o4 = threadIdx.x * 4;
  const float* Qb = Q + (size_t)b * NPTS * O;
  float4 mx = make_float4(-3.4e38f, -3.4e38f, -3.4e38f, -3.4e38f);
  #pragma unroll 4
  for (int k = 0; k < KNN; ++k) {
    const float4 q = *(const float4*)(Qb + (size_t)sidx[k] * O + o4);
    if (k + 1 < KNN) __builtin_prefetch(Qb + (size_t)sidx[k + 1] * O + o4, 0, 0);
    mx.x = fmaxf(mx.x, q.x); mx.y = fmaxf(mx.y, q.y);
    mx.z = fmaxf(mx.z, q.z); mx.w = fmaxf(mx.w, q.w);
  }
  const float4 p = *(const float4*)(P + ((size_t)b * NPTS + n) * O + o4);
  float* outp = xf + (size_t)b * CH_TOT * NPTS + (size_t)(chan_off + o4) * NPTS + n;
  float vals[4] = {p.x + mx.x, p.y + mx.y, p.z + mx.z, p.w + mx.w};
  #pragma unroll
  for (int jj = 0; jj < 4; ++jj) {
    float v = vals[jj];
    v = gamma[o4 + jj] * BN_INVF * v + beta[o4 + jj];
    v = (v >= 0.f) ? v : SLOPEF * v;
    outp[(size_t)jj * NPTS] = v;
  }
}

// ------------------------------------------------------------- final stage
__global__ void zero_kernel(float* __restrict__ p, int nelem) {
  int t = blockIdx.x * blockDim.x + threadIdx.x;
  if (t < nelem) p[t] = 0.f;
}

// 16 rows (o) x 128 cols (n) per wave: 8 accumulators share each A fragment.
__global__ void final_kernel(const float* __restrict__ xf, const float* __restrict__ wm,
                             const float* __restrict__ bm, const float* __restrict__ gm,
                             const float* __restrict__ betam, float* __restrict__ x4) {
  const int lane = threadIdx.x;
  const int o0 = blockIdx.x * 16;
  const int b  = blockIdx.z;
  const int m = lane & 15;
  const int half = (lane >> 4) & 1;
  const int n0 = blockIdx.y * 128;   // 8 tiles of 16 columns
  const float* xfb = xf + (size_t)b * CH_TOT * NPTS;
  v8f acc[8];
  #pragma unroll
  for (int t = 0; t < 8; ++t) acc[t] = (v8f)V8F_ZERO;
  for (int c0 = 0; c0 < CH_TOT; c0 += 4) {
    const int k0 = c0 + 2 * half;
    v2f a;
    a.x = wm[(size_t)(o0 + m) * CH_TOT + k0];
    a.y = wm[(size_t)(o0 + m) * CH_TOT + k0 + 1];
    const float* p0 = xfb + (size_t)k0       * NPTS + n0 + m;
    const float* p1 = xfb + (size_t)(k0 + 1) * NPTS + n0 + m;
    #pragma unroll
    for (int t = 0; t < 8; ++t) {
      v2f bb;
      bb.x = p0[t * 16];
      bb.y = p1[t * 16];
      acc[t] = wmma_f32_16x16x4(a, bb, acc[t]);
    }
  }
  #pragma unroll
  for (int r = 0; r < 8; ++r) {
    const int row = half * 8 + r;
    const float bmv = bm[o0 + row];
    const float gmv = gm[o0 + row] * BN_INVF;
    const float btv = betam[o0 + row];
    float runmax = 0.f;                       // ReLU output >= 0
    #pragma unroll
    for (int t = 0; t < 8; ++t) {
      float v = acc[t][r] + bmv;
      v = fmaxf(gmv * v + btv, 0.f);
      runmax = fmaxf(runmax, v);
    }
    #pragma unroll
    for (int off = 1; off < 16; off <<= 1)
      runmax = fmaxf(runmax, __shfl_xor(runmax, off, 32));
    if ((lane & 15) == 0) {
      // non-negative floats: IEEE order == unsigned-int order
      atomicMax((unsigned int*)(x4 + (size_t)b * 1024 + o0 + row), __float_as_uint(runmax));
    }
  }
}

// ------------------------------------------------------------------ driver
template <int C, int O>
static void run_layer(const float* xin, long long bstride,
                      const float* w, const float* gamma, const float* beta,
                      int chan_off, float* xf,
                      float* xx, float* dist, int* idx, float* P, float* Q,
                      hipStream_t stream) {
  xx_kernel<<<(BATCH * NPTS) / 256, 256, 0, stream>>>(xin, bstride, C, xx);
  for (int b = 0; b < BATCH; ++b) {
    gram_kernel<C><<<dim3(NPTS / 16, NPTS / 16, 1), 32, 0, stream>>>(
        xin + (size_t)b * (size_t)bstride, xx + (size_t)b * NPTS, dist);
    topk_kernel<<<NPTS, 32, 0, stream>>>(dist, idx + (size_t)b * NPTS * KNN);
  }
  pq_kernel<C, O><<<dim3(NPTS / 16, O / 16, BATCH), 32, 0, stream>>>(
      xin, bstride, w, P, Q);
  edge_max_kernel<O><<<dim3(NPTS, BATCH), O / 4, 0, stream>>>(
      P, Q, idx, gamma, beta, xf, chan_off);
}

extern "C" void kernel_launch(void* const* d_in, const int* in_sizes, int n_in,
                              void* d_out, int out_size, void* d_ws, size_t ws_size,
                              hipStream_t stream) {
  const float* x      = (const float*)d_in[0];
  const float* w1     = (const float*)d_in[1];
  const float* gamma1 = (const float*)d_in[2];
  const float* beta1  = (const float*)d_in[3];
  const float* w2     = (const float*)d_in[4];
  const float* gamma2 = (const float*)d_in[5];
  const float* beta2  = (const float*)d_in[6];
  const float* w3     = (const float*)d_in[7];
  const float* gamma3 = (const float*)d_in[8];
  const float* beta3  = (const float*)d_in[9];
  const float* wm     = (const float*)d_in[10];
  const float* bm     = (const float*)d_in[11];
  const float* gm     = (const float*)d_in[12];
  const float* betam  = (const float*)d_in[13];

  float* x4 = (float*)d_out;                       // (8, 1024)
  float* xf = (float*)d_out + BATCH * 1024;        // (8, 448, 2048)

  // workspace layout (~51 MiB total)
  char* ws = (char*)d_ws;
  size_t off = 0;
  float* xx   = (float*)(ws + off); off += (size_t)BATCH * NPTS * 4;            // 64 KB
  off = (off + 255) & ~(size_t)255;
  float* dist = (float*)(ws + off); off += (size_t)NPTS * NPTS * 4;             // 16 MiB (per batch)
  off = (off + 255) & ~(size_t)255;
  int*   idx  = (int*)  (ws + off); off += (size_t)BATCH * NPTS * KNN * 4;      // 2.5 MiB
  off = (off + 255) & ~(size_t)255;
  float* P    = (float*)(ws + off); off += (size_t)BATCH * NPTS * 256 * 4;      // 16 MiB
  off = (off + 255) & ~(size_t)255;
  float* Q    = (float*)(ws + off);                                             // 16 MiB

  // EdgeConv 1: C=3 -> O=64, channels [0,64)
  run_layer<3, 64>(x, (long long)3 * NPTS, w1, gamma1, beta1, 0,
                   xf, xx, dist, idx, P, Q, stream);
  // EdgeConv 2: input = x1 (xf channels 0..63), C=64 -> O=128, channels [64,192)
  run_layer<64, 128>(xf, (long long)CH_TOT * NPTS, w2, gamma2, beta2, 64,
                     xf, xx, dist, idx, P, Q, stream);
  // EdgeConv 3: input = x2 (xf channels 64..191), C=128 -> O=256, channels [192,448)
  run_layer<128, 256>(xf + (size_t)64 * NPTS, (long long)CH_TOT * NPTS, w3, gamma3, beta3, 192,
                      xf, xx, dist, idx, P, Q, stream);

  // Final fused GEMM + BN + ReLU + max over N
  zero_kernel<<<(BATCH * 1024 + 255) / 256, 256, 0, stream>>>(x4, BATCH * 1024);
  final_kernel<<<dim3(1024 / 16, 16, BATCH), 32, 0, stream>>>(xf, wm, bm, gm, betam, x4);
}